// RandScatter_27797028339997
// MI455X (gfx1250) — compile-verified
//
#include <hip/hip_runtime.h>
#include <hip/hip_bf16.h>
#include <cstdint>

// Problem shape (fixed by the reference): N=4096 tokens, D=1024, P=4 paths.
#define N_TOK 4096
#define DIM   1024
#define PATHS 4

// ---------------------------------------------------------------------------
// CDNA5 async global<->LDS DMA helpers (ASYNCcnt-tracked, gfx1250).
// Per CDNA5 ISA: load:  VDST = LDS-offset VGPR, then 64b addr.
//                             store: 64b addr, then LDS-offset VGPR.
// LDS offset = low 32 bits of the generic pointer (flat->LDS truncation rule).
// ---------------------------------------------------------------------------
__device__ __forceinline__ void async_load_b128(const void* gsrc, void* lds) {
  uint32_t loff = (uint32_t)(uintptr_t)lds;
  unsigned long long ga = (unsigned long long)(uintptr_t)gsrc;
  asm volatile("global_load_async_to_lds_b128 %0, %1, off"
               :: "v"(loff), "v"(ga) : "memory");
}

__device__ __forceinline__ void async_store_b128(void* gdst, const void* lds) {
  uint32_t loff = (uint32_t)(uintptr_t)lds;
  unsigned long long ga = (unsigned long long)(uintptr_t)gdst;
  asm volatile("global_store_async_from_lds_b128 %0, %1, off"
               :: "v"(ga), "v"(loff) : "memory");
}

__device__ __forceinline__ void wait_async0() {
#if __has_builtin(__builtin_amdgcn_s_wait_asynccnt)
  __builtin_amdgcn_s_wait_asynccnt(0);
#else
  asm volatile("s_wait_asynccnt 0" ::: "memory");
#endif
}

// ---------------------------------------------------------------------------
// Kernel 1: routing. Single block of 1024 threads; each owns 4 tokens.
// Produces: route[N] (i32 into d_out), counts[P] (i32 into d_out),
//           token_for[P*N] inverse map in d_ws (only slots < counts[p] valid).
// ---------------------------------------------------------------------------
__global__ __launch_bounds__(1024)
void route_kernel(const float* __restrict__ score,   // [N_TOK, PATHS]
                  int* __restrict__ token_for,       // d_ws, [PATHS * N_TOK]
                  int* __restrict__ counts_out,      // [PATHS]
                  int* __restrict__ route_out) {     // [N_TOK]
  __shared__ unsigned long long scan[1024];
  const int t = threadIdx.x;

  int routes[4];
  unsigned long long packed = 0ull;  // 4 x 16-bit per-path local counts
#pragma unroll
  for (int j = 0; j < 4; ++j) {
    const int n = t * 4 + j;
    const float4 s = reinterpret_cast<const float4*>(score)[n];
    int r = 0;
    float best = s.x;
    if (s.y > best) { best = s.y; r = 1; }   // strict >: first-max, like argmax
    if (s.z > best) { best = s.z; r = 2; }
    if (s.w > best) { best = s.w; r = 3; }
    routes[j] = r;
    packed += 1ull << (16 * r);
  }

  scan[t] = packed;
  __syncthreads();
  // Hillis-Steele inclusive scan over 1024 packed count vectors.
  for (int off = 1; off < 1024; off <<= 1) {
    unsigned long long v = (t >= off) ? scan[t - off] : 0ull;
    __syncthreads();
    scan[t] += v;
    __syncthreads();
  }
  const unsigned long long total = scan[1023];
  const unsigned long long excl  = scan[t] - packed;  // exclusive prefix

  if (t < PATHS) counts_out[t] = (int)((total >> (16 * t)) & 0xFFFFull);

  int running[PATHS] = {0, 0, 0, 0};
#pragma unroll
  for (int j = 0; j < 4; ++j) {
    const int n = t * 4 + j;
    const int r = routes[j];
    const int pos = (int)((excl >> (16 * r)) & 0xFFFFull) + running[r];
    running[r]++;
    route_out[n] = r;
    token_for[r * N_TOK + pos] = n;
  }
}

// ---------------------------------------------------------------------------
// Kernel 2: dispatch. One block per output row (p, slot). 256 threads x 16 B.
// Valid rows: async DMA global -> LDS -> global (ASYNCcnt path).
// Padding rows: store zeros. Exactly one 4 KB store per row (64 MB total).
// ---------------------------------------------------------------------------
__global__ __launch_bounds__(256)
void dispatch_kernel(const float* __restrict__ inputs,   // [N_TOK, DIM]
                     const int* __restrict__ token_for,  // [PATHS * N_TOK]
                     const int* __restrict__ counts,     // [PATHS]
                     float* __restrict__ out) {          // [PATHS, N_TOK, DIM]
  __shared__ __align__(16) float lds[DIM];  // 4 KB staging buffer
  const int b    = blockIdx.x;              // 0 .. PATHS*N_TOK-1
  const int p    = b >> 12;                 // b / N_TOK
  const int slot = b & (N_TOK - 1);
  const int tid  = threadIdx.x;

  float* dst = out + (size_t)b * DIM;

  if (slot < counts[p]) {                   // uniform per-block branch
    const int tok = token_for[b];
    const float* src = inputs + (size_t)tok * DIM;
    // Lane-exclusive 16B slots: global -> LDS (async)
    async_load_b128(src + 4 * tid, &lds[4 * tid]);
    wait_async0();                          // this wave's loads complete
    // LDS -> global (async); s_endpgm's implicit wait-idle drains the rest
    async_store_b128(dst + 4 * tid, &lds[4 * tid]);
  } else {
    float4 z = make_float4(0.f, 0.f, 0.f, 0.f);
    reinterpret_cast<float4*>(dst)[tid] = z;
  }
}

// ---------------------------------------------------------------------------
extern "C" void kernel_launch(void* const* d_in, const int* in_sizes, int n_in,
                              void* d_out, int out_size, void* d_ws, size_t ws_size,
                              hipStream_t stream) {
  const float* inputs = (const float*)d_in[0];   // [N_TOK, DIM] f32
  const float* score  = (const float*)d_in[1];   // [N_TOK, PATHS] f32

  float* out = (float*)d_out;
  // d_out layout (flat, return order): dispatched | counts | route
  int* counts_out = (int*)(out + (size_t)PATHS * N_TOK * DIM);
  int* route_out  = counts_out + PATHS;

  int* token_for = (int*)d_ws;                   // PATHS * N_TOK ints (64 KB)

  route_kernel<<<1, 1024, 0, stream>>>(score, token_for, counts_out, route_out);
  dispatch_kernel<<<PATHS * N_TOK, 256, 0, stream>>>(inputs, token_for,
                                                     counts_out, out);
}